// CAConvV2_50998441672737
// MI455X (gfx1250) — compile-verified
//
#include <hip/hip_runtime.h>
#include <hip/hip_bf16.h>

// CAConvV2 on gfx1250 (MI455X).
// out[f, t, j=l*4+m, i] = (t >= i+l) ? bias[o] + sum_k w[o,k]*x[m, t-2+k, fc] : 0
//   with fc = (f - (i+l)) mod 512, o = fc*24 + i*3 + l.
// Store-bandwidth bound (201 MB out vs 8 MB in, ~8.6 us floor at 23.3 TB/s):
//  - Phase 1: x halo tile -> LDS via GLOBAL_LOAD_ASYNC_TO_LDS_B32 (ASYNCcnt).
//  - Phase 2: grouped 3-tap conv as V_WMMA_F32_16X16X4_F32, D tiles -> LDS.
//  - Phase 3: feature-roll resolved from LDS (constant-offset gathers),
//    non-temporal coalesced b128 stores; time-mask only on the t0==0 row.
// FB=8 / G=17 keeps LDS ~109 KB -> 2 blocks per WGP for store latency hiding.

typedef float v2f __attribute__((ext_vector_type(2)));
typedef float v4f __attribute__((ext_vector_type(4)));
typedef float v8f __attribute__((ext_vector_type(8)));

typedef __attribute__((address_space(1))) int gas_i32;   // global AS
typedef __attribute__((address_space(3))) int las_i32;   // LDS AS

#define F_  512
#define T_  1024
#define M_  4
#define CH  24     // K*L channels per group
#define FB  8      // output features per block
#define TT  16     // time steps per block
#define G_  17     // halo groups: fc in [fb-9, fb+7]
#define YSTRIDE (M_ * TT * CH)          // 1536 words per halo group
#define DSTEP   (YSTRIDE - 3)           // 1533: delta between d and d+1 gathers

#if defined(__has_builtin)
#if __has_builtin(__builtin_amdgcn_global_load_async_to_lds_b32)
#define HAVE_ASYNC_LDS 1
#endif
#endif

__global__ __launch_bounds__(256)
void CAConvV2_kernel(const float* __restrict__ x,
                     const float* __restrict__ w,
                     const float* __restrict__ bias,
                     float* __restrict__ out) {
  __shared__ float xls[M_][TT + 2][G_];        //  4.9 KB  x halo tile
  __shared__ float yls[G_][M_][TT][CH];        // 104.4 KB conv results

  const int tid = threadIdx.x;
  const int fb  = blockIdx.x * FB;
  const int t0  = blockIdx.y * TT;

  // ---- Phase 1: stage x[m, t0-2 .. t0+15, fc(fb-9..fb+7)] into LDS ----
  for (int idx = tid; idx < M_ * (TT + 2) * G_; idx += 256) {
    int m  = idx / ((TT + 2) * G_);
    int r  = idx % ((TT + 2) * G_);
    int tt = r / G_;
    int g  = r % G_;
    int t  = t0 + tt - 2;
    int fc = (fb - 9 + g + F_) & (F_ - 1);
#ifdef HAVE_ASYNC_LDS
    if (t >= 0) {
      __builtin_amdgcn_global_load_async_to_lds_b32(
          (gas_i32*)(x + ((size_t)m * T_ + t) * F_ + fc),
          (las_i32*)&xls[m][tt][g],
          /*offset=*/0, /*cpol=*/0);
    } else {
      xls[m][tt][g] = 0.0f;
    }
#else
    float v = 0.0f;
    if (t >= 0) v = x[((size_t)m * T_ + t) * F_ + fc];
    xls[m][tt][g] = v;
#endif
  }
#ifdef HAVE_ASYNC_LDS
#if __has_builtin(__builtin_amdgcn_s_wait_asynccnt)
  __builtin_amdgcn_s_wait_asynccnt(0);
#else
  asm volatile("s_wait_asynccnt 0x0" ::: "memory");
#endif
#endif
  __syncthreads();

  // ---- Phase 2: grouped 3-tap conv via V_WMMA_F32_16X16X4_F32 ----
  // One WMMA per (group g, batch m, 16-channel tile): D[t][n] =
  //   sum_{k<3} x(t0+t-2+k)*w[o_n,k] + bias[o_n],  o_n = fc*24 + n.
  // 32-bit A (16x4) lane layout: lanes 0-15 row t=lh hold K=0/1 in v0/v1,
  // lanes 16-31 hold K=2/3.  B (4x16) mirrors: lanes 0-15 rows K=0/1,
  // lanes 16-31 rows K=2/3, column N = lane&15.
  const int wave = tid >> 5;
  const int lane = tid & 31;
  const int hl   = lane >> 4;   // which 16-lane half
  const int lh   = lane & 15;

  for (int task = wave; task < G_ * M_ * 2; task += 8) {   // 136/8 = 17 each
    int g     = task >> 3;
    int rem   = task & 7;
    int m     = rem >> 1;
    int ntile = rem & 1;
    int fc    = (fb - 9 + g + F_) & (F_ - 1);
    int n     = ntile * 16 + lh;
    bool vn   = (n < CH);
    int o     = fc * CH + n;

    float b0 = 0.0f, b1 = 0.0f, cb = 0.0f;
    if (vn) {
      if (hl == 0) { b0 = w[o * 3 + 0]; b1 = w[o * 3 + 1]; }   // rows K=0,1
      else         { b0 = w[o * 3 + 2]; }                      // rows K=2,3(=0)
      cb = bias[o];
    }
    // A operand: taps of x for row t = lh (upper half rows handled by HW)
    float a0 = xls[m][lh + (hl ? 2 : 0)][g];
    float a1 = hl ? 0.0f : xls[m][lh + 1][g];

    v2f A = {a0, a1};
    v2f B = {b0, b1};
    v8f C;
    for (int r2 = 0; r2 < 8; ++r2) C[r2] = cb;

    v8f D = __builtin_amdgcn_wmma_f32_16x16x4_f32(
        /*neg_a=*/false, A, /*neg_b=*/false, B,
        /*c_mod=*/(short)0, C, /*reuse_a=*/false, /*reuse_b=*/false);

    // D layout: VGPR r -> (t = r + 8*hl, n = lane&15 (+ntile*16))
    if (vn) {
      for (int r2 = 0; r2 < 8; ++r2)
        yls[g][m][hl * 8 + r2][n] = D[r2];
    }
  }
  __syncthreads();

  // ---- Phase 3: coalesced non-temporal b128 store of the block tile ----
  // Element e = 4*(tid + 256*it) over FB*TT*96 = 12288 elements; per
  // iteration e advances +1024 (= +10 rows, +64 in ji mod 96); maintain
  // (row, ji) incrementally.  The four gathered yls words sit at constant
  // deltas of DSTEP words (g -1, n +3), so one base address + immediate
  // offsets covers the whole float4.  Lanes stay byte-consecutive ->
  // full 128B-per-wave store lines.
  {
    const float* yflat = &yls[0][0][0][0];
    int base = tid * 4;            // e at it = 0 (0..1020)
    int row  = base / 96;          // one division per thread, total
    int ji   = base - row * 96;

    if (t0 != 0) {
      // Common path (63/64 blocks): t = t0+tl >= 16 > 9 >= s, no time-mask.
#pragma unroll 4
      for (int it = 0; it < 12; ++it) {
        int fl    = row >> 4;
        int tl    = row & 15;
        int j     = ji >> 3;
        int ibase = ji & 7;        // 0 or 4
        int l     = j >> 2;
        int m     = j & 3;
        // base gather address for d = 3 (largest s -> smallest g)
        int g3 = fl + 9 - (ibase + 3 + l);
        int a3 = ((g3 * M_ + m) * TT + tl) * CH + (ibase + 3) * 3 + l;
        v4f v;
        v[3] = yflat[a3];
        v[2] = yflat[a3 + DSTEP];
        v[1] = yflat[a3 + 2 * DSTEP];
        v[0] = yflat[a3 + 3 * DSTEP];
        unsigned int oidx =
            (((unsigned int)(fb + fl) * T_ + (unsigned int)(t0 + tl)) * 12u
             + (unsigned int)j) * 8u + (unsigned int)ibase;
        __builtin_nontemporal_store(v, (v4f*)(out + oidx));
        ji += 64; row += 10;
        if (ji >= 96) { ji -= 96; row += 1; }
      }
    } else {
      // t0 == 0: apply the t >= s causal mask via select (loads unconditional).
#pragma unroll 4
      for (int it = 0; it < 12; ++it) {
        int fl    = row >> 4;
        int tl    = row & 15;
        int j     = ji >> 3;
        int ibase = ji & 7;
        int l     = j >> 2;
        int m     = j & 3;
        int g3 = fl + 9 - (ibase + 3 + l);
        int a3 = ((g3 * M_ + m) * TT + tl) * CH + (ibase + 3) * 3 + l;
        v4f v;
        for (int d = 0; d < 4; ++d) {
          int s = ibase + d + l;
          float val = yflat[a3 + (3 - d) * DSTEP];
          v[d] = val * ((tl >= s) ? 1.0f : 0.0f);
        }
        unsigned int oidx =
            (((unsigned int)(fb + fl) * T_ + (unsigned int)tl) * 12u
             + (unsigned int)j) * 8u + (unsigned int)ibase;
        __builtin_nontemporal_store(v, (v4f*)(out + oidx));
        ji += 64; row += 10;
        if (ji >= 96) { ji -= 96; row += 1; }
      }
    }
  }
}

extern "C" void kernel_launch(void* const* d_in, const int* in_sizes, int n_in,
                              void* d_out, int out_size, void* d_ws, size_t ws_size,
                              hipStream_t stream) {
  const float* x    = (const float*)d_in[0];
  const float* w    = (const float*)d_in[1];
  const float* bias = (const float*)d_in[2];
  float* out = (float*)d_out;
  (void)in_sizes; (void)n_in; (void)out_size; (void)d_ws; (void)ws_size;

  dim3 grid(F_ / FB, T_ / TT);   // 64 x 64 blocks
  dim3 block(256);               // 8 wave32
  CAConvV2_kernel<<<grid, block, 0, stream>>>(x, w, bias, out);
}